// Attention_76347338653911
// MI455X (gfx1250) — compile-verified
//
#include <hip/hip_runtime.h>
#include <hip/hip_bf16.h>
#include <math.h>

// ---------------- problem constants ----------------
#define B_   64
#define N_   196
#define C_   768
#define H_   12
#define HD   64
#define P_   196
#define BN_EPS 1e-5f
#define SCALE_ 0.125f   // hd^-0.5 = 1/8

// workspace element counts (floats)
#define QVSZ ((size_t)B_ * H_ * N_ * HD)      // 9,633,792
#define ASZ  ((size_t)B_ * H_ * N_ * P_)      // 29,503,488

// stats layout (floats)
#define ST_ASUM  0
#define ST_ASQ   12
#define ST_VSUM  24
#define ST_VSQ   36
#define ST_AMEAN 48
#define ST_ARSTD 60
#define ST_VMEAN 72
#define ST_VRSTD 84
#define ST_TOTAL 96

typedef __attribute__((ext_vector_type(2))) float v2f;
typedef __attribute__((ext_vector_type(8))) float v8f;

// fp32 matrix core: D = A(16x4) * B(4x16) + C, exact fp32 accumulate.
__device__ __forceinline__ v8f wmma4(v2f a, v2f b, v8f c) {
    return __builtin_amdgcn_wmma_f32_16x16x4_f32(
        false, a, false, b, (short)0, c, false, false);
}

__device__ __forceinline__ v8f vzero8() {
    v8f z = {0.f, 0.f, 0.f, 0.f, 0.f, 0.f, 0.f, 0.f};
    return z;
}

__device__ __forceinline__ float wave_sum(float x) {
    for (int off = 16; off; off >>= 1) x += __shfl_xor(x, off, 32);
    return x;
}
__device__ __forceinline__ float wave_max(float x) {
    for (int off = 16; off; off >>= 1) x = fmaxf(x, __shfl_xor(x, off, 32));
    return x;
}

// ---------------- async global->LDS staging (CDNA5 path, guarded) ----------------
#if __has_builtin(__builtin_amdgcn_global_load_async_to_lds_b128) && \
    __has_builtin(__builtin_amdgcn_s_wait_asynccnt)
#define ASYNC_LDS 1
typedef int v4i_ __attribute__((vector_size(16)));
typedef v4i_ __attribute__((address_space(1)))* gas_ptr;   // global 16B vector ptr
typedef v4i_ __attribute__((address_space(3)))* las_ptr;   // LDS    16B vector ptr
#else
#define ASYNC_LDS 0
#endif

// copy nrows x ncols floats (ncols % 4 == 0) global->LDS with row padding.
// dst rows must be 16B aligned (dst_ld % 4 == 0).
__device__ __forceinline__ void stage_tile(const float* __restrict__ src, int src_ld,
                                           float* __restrict__ dst, int dst_ld,
                                           int nrows, int ncols, int tid, int nthr) {
    const int cpr = ncols >> 2;             // float4 chunks per row
    const int total = nrows * cpr;
    for (int i = tid; i < total; i += nthr) {
        const int r = i / cpr;
        const int c = i - r * cpr;
        const float* s = src + (size_t)r * src_ld + 4 * c;
        float* d = dst + r * dst_ld + 4 * c;
#if ASYNC_LDS
        __builtin_amdgcn_global_load_async_to_lds_b128((gas_ptr)s, (las_ptr)d, 0, 0);
#else
        *(float4*)d = *(const float4*)s;
#endif
    }
}

__device__ __forceinline__ void stage_wait() {
#if ASYNC_LDS
    __builtin_amdgcn_s_wait_asynccnt(0);
#endif
    __syncthreads();
}

// ---------------- kernel 1: zero stats scratch ----------------
__global__ void k_zero(float* __restrict__ stats) {
    int t = threadIdx.x;
    if (t < ST_TOTAL) stats[t] = 0.f;
}

// ---------------- kernel 2: qv = x @ W_qv ----------------
// grid (784, 3), block 256. Block stages its 16x768 A panel in LDS (async),
// each wave computes a 16x64 slab (4 accumulators) of a 64-aligned column group,
// scatters into q/v [B,H,N,hd] and fuses the v-BN sum/sumsq reduction.
#define QV_LDA 772   // 768 + 4 pad: stride % 64 == 4 -> conflict-free ds_load_b64
__global__ void __launch_bounds__(256) k_qv(
        const float* __restrict__ x, const float* __restrict__ Wqv,
        float* __restrict__ q, float* __restrict__ v, float* __restrict__ stats) {
    __shared__ float sa[16 * QV_LDA];

    const int wave = threadIdx.x >> 5;
    const int lane = threadIdx.x & 31;
    const int lane16 = lane & 15;
    const int khalf  = lane >> 4;
    const int m0 = blockIdx.x * 16;

    stage_tile(x + (size_t)m0 * C_, C_, sa, QV_LDA, 16, C_, threadIdx.x, 256);
    stage_wait();

    const int n0 = (blockIdx.y * 8 + wave) * 64;      // 64-aligned column group
    const int ldb = 2 * C_;

    v8f acc[4] = {vzero8(), vzero8(), vzero8(), vzero8()};
    for (int k0 = 0; k0 < C_; k0 += 4) {
        const int kk = k0 + 2 * khalf;
        const v2f a = *(const v2f*)&sa[lane16 * QV_LDA + kk];
        const float* bp = Wqv + (size_t)kk * ldb + n0 + lane16;
        __builtin_prefetch(bp + 8 * ldb, 0, 1);
#pragma unroll
        for (int j = 0; j < 4; ++j) {
            v2f b;
            b.x = bp[16 * j];
            b.y = bp[ldb + 16 * j];
            acc[j] = wmma4(a, b, acc[j]);
        }
    }

    // epilogue: group is entirely q or entirely v, and within one head (64 | n0)
    const bool is_v = (n0 >= C_);
    const int h = (n0 - (is_v ? C_ : 0)) / HD;
    float* dst = is_v ? v : q;

    float lsum = 0.f, lsq = 0.f;
#pragma unroll
    for (int r = 0; r < 8; ++r) {
        const int row = m0 + r + 8 * khalf;           // < 12544 always
        const int bb  = row / N_;
        const int nn  = row - bb * N_;
        float* base = dst + (((size_t)bb * H_ + h) * N_ + nn) * HD + lane16;
#pragma unroll
        for (int j = 0; j < 4; ++j) {
            const float val = acc[j][r];
            base[16 * j] = val;
            lsum += val; lsq += val * val;
        }
    }
    if (is_v) {                                       // h uniform across the wave
        lsum = wave_sum(lsum);
        lsq  = wave_sum(lsq);
        if (lane == 0) {
            atomicAdd(&stats[ST_VSUM + h], lsum);
            atomicAdd(&stats[ST_VSQ  + h], lsq);
        }
    }
}

// ---------------- kernel 3: attn[b,h] = q[b,h] @ k_ext^T + fused BN stats ----------
// grid B*H=768 blocks. k_ext (196x64) staged in LDS once; waves compute 16x64 slabs.
#define SK_LD 68     // 64 + 4 pad
__global__ void __launch_bounds__(256) k_scores(
        const float* __restrict__ q, const float* __restrict__ kext,
        float* __restrict__ attn, float* __restrict__ stats) {
    __shared__ float sk[P_ * SK_LD];
    __shared__ float ssum[8], ssq[8];

    const int bh = blockIdx.x;
    const int h  = bh % H_;
    const int wave = threadIdx.x >> 5;
    const int lane = threadIdx.x & 31;
    const int lane16 = lane & 15;
    const int khalf  = lane >> 4;

    stage_tile(kext, HD, sk, SK_LD, P_, HD, threadIdx.x, 256);
    stage_wait();

    const float* A  = q    + (size_t)bh * N_ * HD;    // [196, 64]
    float*       Cm = attn + (size_t)bh * N_ * P_;    // [196, 196]

    float wsum = 0.f, wsq = 0.f;
    for (int t = wave; t < 13 * 4; t += 8) {          // 13 m-tiles x 4 p-groups
        const int m0 = (t >> 2) * 16;
        const int n0 = (t & 3) * 64;
        const int m = m0 + lane16;
        const bool mv = (m < N_);

        v8f acc[4] = {vzero8(), vzero8(), vzero8(), vzero8()};
        for (int k0 = 0; k0 < HD; k0 += 4) {
            const int kk = k0 + 2 * khalf;
            v2f a = {0.f, 0.f};
            if (mv) { const float* ap = A + (size_t)m * HD + kk; a.x = ap[0]; a.y = ap[1]; }
#pragma unroll
            for (int j = 0; j < 4; ++j) {
                const int pcol = n0 + 16 * j + lane16;
                v2f b = {0.f, 0.f};
                if (pcol < P_) b = *(const v2f*)&sk[pcol * SK_LD + kk];
                acc[j] = wmma4(a, b, acc[j]);
            }
        }
#pragma unroll
        for (int r = 0; r < 8; ++r) {
            const int row = m0 + r + 8 * khalf;
            if (row < N_) {
                float* base = Cm + (size_t)row * P_;
#pragma unroll
                for (int j = 0; j < 4; ++j) {
                    const int pcol = n0 + 16 * j + lane16;
                    if (pcol < P_) {
                        const float val = acc[j][r];
                        base[pcol] = val;
                        wsum += val; wsq += val * val;
                    }
                }
            }
        }
    }

    wsum = wave_sum(wsum);
    wsq  = wave_sum(wsq);
    if (lane == 0) { ssum[wave] = wsum; ssq[wave] = wsq; }
    __syncthreads();
    if (threadIdx.x == 0) {
        float ts = 0.f, tq = 0.f;
#pragma unroll
        for (int i = 0; i < 8; ++i) { ts += ssum[i]; tq += ssq[i]; }
        atomicAdd(&stats[ST_ASUM + h], ts);
        atomicAdd(&stats[ST_ASQ  + h], tq);
    }
}

// ---------------- kernel 4: finalize BN stats ----------------
__global__ void k_finalize(float* __restrict__ stats) {
    const int t = threadIdx.x;
    if (t < H_) {
        const float cnt = (float)B_ * (float)N_ * (float)P_;
        const float mean = stats[ST_ASUM + t] / cnt;
        const float var  = stats[ST_ASQ + t] / cnt - mean * mean;
        stats[ST_AMEAN + t] = mean;
        stats[ST_ARSTD + t] = rsqrtf(var + BN_EPS);
    } else if (t >= 32 && t < 32 + H_) {
        const int h = t - 32;
        const float cnt = (float)B_ * (float)N_ * (float)HD;
        const float mean = stats[ST_VSUM + h] / cnt;
        const float var  = stats[ST_VSQ + h] / cnt - mean * mean;
        stats[ST_VMEAN + h] = mean;
        stats[ST_VRSTD + h] = rsqrtf(var + BN_EPS);
    }
}

// ---------------- kernel 5: row softmax with folded BN affine + bias + scale ------
// one wave per row of 196; statically-indexed register tile (no scratch).
__global__ void __launch_bounds__(256) k_softmax(
        float* __restrict__ attn, const float* __restrict__ bias,
        const float* __restrict__ gamma, const float* __restrict__ beta,
        const float* __restrict__ stats) {
    const int wave = threadIdx.x >> 5;
    const int lane = threadIdx.x & 31;
    const long long row = (long long)blockIdx.x * 8 + wave;   // 0 .. B*H*N-1
    const int h = (int)((row / N_) % H_);

    const float rstd = stats[ST_ARSTD + h];
    const float g    = gamma[h] * rstd;
    const float bt   = beta[h] - stats[ST_AMEAN + h] * g;
    float* rp = attn + (size_t)row * P_;

    float vals[7];
    float mx = -1e30f;
#pragma unroll
    for (int i = 0; i < 7; ++i) {
        const int p = lane + 32 * i;
        float vv = -1e30f;
        if (p < P_) vv = (rp[p] * g + bt + bias[p]) * SCALE_;
        vals[i] = vv;
        mx = fmaxf(mx, vv);
    }
    mx = wave_max(mx);
    float sum = 0.f;
#pragma unroll
    for (int i = 0; i < 7; ++i) {
        const int p = lane + 32 * i;
        const float e = (p < P_) ? __expf(vals[i] - mx) : 0.f;
        vals[i] = e;
        sum += e;
    }
    sum = wave_sum(sum);
    const float inv = 1.0f / sum;
#pragma unroll
    for (int i = 0; i < 7; ++i) {
        const int p = lane + 32 * i;
        if (p < P_) rp[p] = vals[i] * inv;
    }
}

// ---------------- kernel 6: out = attn @ BN(v), transposed store to [B,N,C] -------
// grid B*H=768. BN(v) staged (pre-normalized) in LDS; waves compute 16x64 slabs.
__global__ void __launch_bounds__(256) k_av(
        const float* __restrict__ attn, const float* __restrict__ v,
        const float* __restrict__ gamma, const float* __restrict__ beta,
        const float* __restrict__ stats, float* __restrict__ o) {
    __shared__ float sv[P_ * SK_LD];                  // BN(v): [p][d] padded

    const int bh = blockIdx.x;
    const int b  = bh / H_;
    const int h  = bh % H_;
    const int wave = threadIdx.x >> 5;
    const int lane = threadIdx.x & 31;
    const int lane16 = lane & 15;
    const int khalf  = lane >> 4;

    const float* A = attn + (size_t)bh * N_ * P_;     // [196, 196] probabilities
    const float* V = v    + (size_t)bh * N_ * HD;     // [196, 64]
    const float rstd = stats[ST_VRSTD + h];
    const float g    = gamma[h] * rstd;
    const float bt   = beta[h] - stats[ST_VMEAN + h] * g;

    for (int i = threadIdx.x; i < P_ * HD; i += 256) {
        const int r = i >> 6, c = i & 63;
        sv[r * SK_LD + c] = V[i] * g + bt;            // BN applied once here
    }
    __syncthreads();

    for (int t = wave; t < 13; t += 8) {              // 13 n-tiles, full 64-wide d
        const int m0 = t * 16;
        const int m = m0 + lane16;
        const bool mv = (m < N_);

        v8f acc[4] = {vzero8(), vzero8(), vzero8(), vzero8()};
        for (int k0 = 0; k0 < P_; k0 += 4) {          // 196 % 4 == 0
            const int kk = k0 + 2 * khalf;
            v2f a = {0.f, 0.f};
            if (mv) { const float* ap = A + (size_t)m * P_ + kk; a.x = ap[0]; a.y = ap[1]; }
#pragma unroll
            for (int j = 0; j < 4; ++j) {
                const int dcol = 16 * j + lane16;
                v2f b;
                b.x = sv[kk * SK_LD + dcol];
                b.y = sv[(kk + 1) * SK_LD + dcol];
                acc[j] = wmma4(a, b, acc[j]);
            }
        }
#pragma unroll
        for (int r = 0; r < 8; ++r) {
            const int row = m0 + r + 8 * khalf;
            if (row < N_) {
                float* base = o + ((size_t)b * N_ + row) * C_ + h * HD + lane16;
#pragma unroll
                for (int j = 0; j < 4; ++j) base[16 * j] = acc[j][r];
            }
        }
    }
}

// ---------------- kernel 7: y = o @ W_proj + b_proj ----------------
// grid (784, 3), block 256. A panel staged in LDS; each wave does 16x32 (2 acc).
__global__ void __launch_bounds__(256) k_proj(
        const float* __restrict__ o, const float* __restrict__ W,
        const float* __restrict__ bias, float* __restrict__ y) {
    __shared__ float sa[16 * QV_LDA];

    const int wave = threadIdx.x >> 5;
    const int lane = threadIdx.x & 31;
    const int lane16 = lane & 15;
    const int khalf  = lane >> 4;
    const int m0 = blockIdx.x * 16;

    stage_tile(o + (size_t)m0 * C_, C_, sa, QV_LDA, 16, C_, threadIdx.x, 256);
    stage_wait();

    const int n0 = (blockIdx.y * 8 + wave) * 32;      // 24 groups of 32 columns

    v8f acc[2] = {vzero8(), vzero8()};
    for (int k0 = 0; k0 < C_; k0 += 4) {
        const int kk = k0 + 2 * khalf;
        const v2f a = *(const v2f*)&sa[lane16 * QV_LDA + kk];
        const float* bp = W + (size_t)kk * C_ + n0 + lane16;
        __builtin_prefetch(bp + 8 * C_, 0, 1);
#pragma unroll
        for (int j = 0; j < 2; ++j) {
            v2f b;
            b.x = bp[16 * j];
            b.y = bp[C_ + 16 * j];
            acc[j] = wmma4(a, b, acc[j]);
        }
    }

#pragma unroll
    for (int r = 0; r < 8; ++r) {
        const int row = m0 + r + 8 * khalf;
        float* base = y + (size_t)row * C_ + n0 + lane16;
#pragma unroll
        for (int j = 0; j < 2; ++j) base[16 * j] = acc[j][r] + bias[n0 + 16 * j + lane16];
    }
}

// ---------------- host launch ----------------
extern "C" void kernel_launch(void* const* d_in, const int* in_sizes, int n_in,
                              void* d_out, int out_size, void* d_ws, size_t ws_size,
                              hipStream_t stream) {
    const float* x     = (const float*)d_in[0];
    const float* Wqv   = (const float*)d_in[1];
    const float* kext  = (const float*)d_in[2];
    const float* abias = (const float*)d_in[3];
    const float* gamma = (const float*)d_in[4];
    const float* beta  = (const float*)d_in[5];
    const float* Wp    = (const float*)d_in[6];
    const float* bp    = (const float*)d_in[7];
    float* y = (float*)d_out;

    float* ws    = (float*)d_ws;
    float* q     = ws;                  // [B,H,N,hd]
    float* v     = q + QVSZ;            // [B,H,N,hd]
    float* attn  = v + QVSZ;            // [B,H,N,P]
    float* o     = attn + ASZ;          // [B,N,C]
    float* stats = o + QVSZ;            // 96 floats

    k_zero    <<<1, 128, 0, stream>>>(stats);
    k_qv      <<<dim3(784, 3), 256, 0, stream>>>(x, Wqv, q, v, stats);
    k_scores  <<<768, 256, 0, stream>>>(q, kext, attn, stats);
    k_finalize<<<1, 64, 0, stream>>>(stats);
    k_softmax <<<18816, 256, 0, stream>>>(attn, abias, gamma, beta, stats);
    k_av      <<<768, 256, 0, stream>>>(attn, v, gamma, beta, stats, o);
    k_proj    <<<dim3(784, 3), 256, 0, stream>>>(o, Wp, bp, y);
}